// reaction_diffusion_nature_90872918049155
// MI455X (gfx1250) — compile-verified
//

#include <hip/hip_runtime.h>
#include <math.h>

// Problem constants from the reference: B is hardcoded to 64 (repeat(64,1,1)).
#define BB     64      // batch
#define TILE_V 64      // nodes per LDS tile
#define PAD    65      // LDS row pitch (words) -> conflict-free column access

// ---------------------------------------------------------------------------
// CDNA5 primitives (gfx1250): async global<->LDS copies + no-return f32 atomic
// ---------------------------------------------------------------------------

// Byte offset of a __shared__ object within the workgroup LDS allocation.
__device__ __forceinline__ unsigned lds_off_u32(const void* p) {
    return (unsigned)(unsigned long long)(__attribute__((address_space(3))) const char*)p;
}

// copies 4 bytes from global address base + goff to the LDS address in ldsoff   (ASYNCcnt-tracked, per-lane)
__device__ __forceinline__ void async_load_b32(unsigned ldsoff, unsigned goff, const void* base) {
    asm volatile("global_load_async_to_lds_b32 %0, %1, %2"
                 :: "v"(ldsoff), "v"(goff), "s"(base) : "memory");
}

// copies 4 bytes from the LDS address in ldsoff to global address base + goff   (ASYNCcnt-tracked, per-lane)
__device__ __forceinline__ void async_store_b32(unsigned goff, unsigned ldsoff, void* base) {
    asm volatile("global_store_async_from_lds_b32 %0, %1, %2"
                 :: "v"(goff), "v"(ldsoff), "s"(base) : "memory");
}

__device__ __forceinline__ void wait_async0() {
    asm volatile("s_wait_asynccnt 0x0" ::: "memory");
}

// Fire-and-forget HW f32 atomic add (no return -> STOREcnt, drained at endpgm).
__device__ __forceinline__ void atomic_add_f32(void* base, unsigned byte_off, float v) {
    asm volatile("global_atomic_add_f32 %0, %1, %2"
                 :: "v"(byte_off), "v"(v), "s"(base) : "memory");
}

// ---------------------------------------------------------------------------
// Kernel 1: transpose x[B,N] -> xT[N,B] via 64x64 LDS tile (async copy engine)
// ---------------------------------------------------------------------------
__global__ void k_transpose_x(const float* __restrict__ x, float* __restrict__ xT, int N) {
    __shared__ float tile[BB * PAD];            // [batch_row][node_col], padded
    const int v0 = blockIdx.x * TILE_V;
    const int t  = threadIdx.x;                 // 0..255

    // Load 64x64 tile: element l -> row r = batch, col c = node. Coalesced per row.
    for (int k = 0; k < 16; ++k) {
        int l = t + 256 * k;
        int r = l >> 6, c = l & 63;
        if (v0 + c < N) {
            async_load_b32(lds_off_u32(&tile[r * PAD + c]),
                           (unsigned)(r * N + v0 + c) * 4u, x);
        }
    }
    wait_async0();
    __syncthreads();

    // Store transposed: xT[(v0+vc)*64 + b], b fastest -> coalesced.
    for (int k = 0; k < 16; ++k) {
        int b  = t & 63;
        int vc = (t >> 6) + 4 * k;
        if (v0 + vc < N) {
            async_store_b32((unsigned)((v0 + vc) * BB + b) * 4u,
                            lds_off_u32(&tile[b * PAD + vc]), xT);
        }
    }
    wait_async0();
}

// ---------------------------------------------------------------------------
// Kernel 2: edge scatter (fused with Laplacian diagonal sums).
// 32 lanes per edge, 2 batches per lane: float2 gathers from node-major xT,
// 256B contiguous atomic bursts into node-major msg[N,B]. Lane 0 of each edge
// group also accumulates the diagonal (column-sum) terms.
// Edge streams are read exactly once per call -> non-temporal hints keep them
// from evicting the L2-resident hot set (xT + msg, hammered by the atomics).
// ---------------------------------------------------------------------------
__global__ void k_edges(const int* __restrict__ ei, const int* __restrict__ ej,
                        const float* __restrict__ wr, const float* __restrict__ wd,
                        const float* __restrict__ xT,
                        float* msg_r, float* msg_d,
                        float* col_r, float* col_d, int E) {
    unsigned gid = blockIdx.x * 256u + threadIdx.x;
    unsigned e   = gid >> 5;
    if (e >= (unsigned)E) return;
    unsigned b2  = (gid & 31u) * 2u;            // batches {b2, b2+1}

    int   i   = __builtin_nontemporal_load(ei + e);
    int   j   = __builtin_nontemporal_load(ej + e);
    float wre = __builtin_nontemporal_load(wr + e);
    float wde = __builtin_nontemporal_load(wd + e);

    const float2 xj = *(const float2*)(xT + (size_t)j * BB + b2);
    const float2 xi = *(const float2*)(xT + (size_t)i * BB + b2);

    unsigned off_r = ((unsigned)i * BB + b2) * 4u;   // msg_react row = src i
    unsigned off_d = ((unsigned)j * BB + b2) * 4u;   // msg_diff  row = src j
    atomic_add_f32(msg_r, off_r,      wre * xj.x);
    atomic_add_f32(msg_r, off_r + 4u, wre * xj.y);
    atomic_add_f32(msg_d, off_d,      wde * xi.x);
    atomic_add_f32(msg_d, off_d + 4u, wde * xi.y);

    if ((gid & 31u) == 0u) {                    // one lane per edge: diagonals
        atomic_add_f32(col_r, (unsigned)j * 4u, wre);   // reaction: dst = j
        atomic_add_f32(col_d, (unsigned)i * 4u, wde);   // diffusion: dst = i
    }
}

// ---------------------------------------------------------------------------
// Kernel 3: combine. Async-load contiguous msg tiles [64v x 64b] into padded
// LDS, transpose through LDS, fully coalesced x reads / out writes.
// ---------------------------------------------------------------------------
__global__ void k_final(const float* __restrict__ x,
                        const float* __restrict__ msg_r, const float* __restrict__ msg_d,
                        const float* __restrict__ col_r, const float* __restrict__ col_d,
                        const float* __restrict__ br,    const float* __restrict__ bd,
                        float* __restrict__ out, int N) {
    __shared__ float tr[TILE_V * PAD];
    __shared__ float td[TILE_V * PAD];
    const int v0 = blockIdx.x * TILE_V;
    const int t  = threadIdx.x;

    // msg tiles are contiguous 16KB blocks (node-major) -> perfectly coalesced.
    for (int k = 0; k < 16; ++k) {
        int l = t + 256 * k;                    // element within tile, v-major
        int v = l >> 6, b = l & 63;
        if (v0 + v < N) {
            unsigned goff = (unsigned)((v0 + v) * BB + b) * 4u;
            async_load_b32(lds_off_u32(&tr[v * PAD + b]), goff, msg_r);
            async_load_b32(lds_off_u32(&td[v * PAD + b]), goff, msg_d);
        }
    }
    wait_async0();
    __syncthreads();

    const int c = t & 63;                       // node column
    const int v = v0 + c;
    if (v < N) {
        float cr  = col_r[v];
        float cd  = col_d[v];
        float brv = br[v];
        float bdv = bd[v];
        for (int k = 0; k < 16; ++k) {
            int   b  = (t >> 6) + 4 * k;
            float xv = x[(size_t)b * N + v];
            float r  = fmaf(cr, xv, brv) - tr[c * PAD + b];   // col*x + bias - msg
            float d  = fmaf(cd, xv, bdv) - td[c * PAD + b];
            out[(size_t)b * N + v] = tanhf(r) + d + xv;
        }
    }
}

// ---------------------------------------------------------------------------
// Launch
// ---------------------------------------------------------------------------
extern "C" void kernel_launch(void* const* d_in, const int* in_sizes, int n_in,
                              void* d_out, int out_size, void* d_ws, size_t ws_size,
                              hipStream_t stream) {
    // setup_inputs order: t, input, edge_i, edge_j, weight_react, weight_diff,
    //                     bias_reaction, bias_diffusion
    const float* x  = (const float*)d_in[1];
    const int*   ei = (const int*)d_in[2];
    const int*   ej = (const int*)d_in[3];
    const float* wr = (const float*)d_in[4];
    const float* wd = (const float*)d_in[5];
    const float* br = (const float*)d_in[6];
    const float* bd = (const float*)d_in[7];
    float* out = (float*)d_out;

    const int E = in_sizes[2];
    const int N = in_sizes[6];
    const size_t nb = (size_t)N * BB;

    // Workspace: xT | msg_react | msg_diff | col_react | col_diff  (~38.8 MB)
    float* xT    = (float*)d_ws;
    float* msg_r = xT + nb;
    float* msg_d = msg_r + nb;
    float* col_r = msg_d + nb;
    float* col_d = col_r + N;

    // Accumulators must be zeroed every call (msg_r/msg_d and col_r/col_d are
    // each contiguous).
    hipMemsetAsync(msg_r, 0, 2 * nb * sizeof(float), stream);
    hipMemsetAsync(col_r, 0, 2 * (size_t)N * sizeof(float), stream);

    const int vblocks = (N + TILE_V - 1) / TILE_V;

    k_transpose_x<<<vblocks, 256, 0, stream>>>(x, xT, N);
    k_edges<<<(int)(((size_t)E * 32 + 255) / 256), 256, 0, stream>>>(
        ei, ej, wr, wd, xT, msg_r, msg_d, col_r, col_d, E);
    k_final<<<vblocks, 256, 0, stream>>>(x, msg_r, msg_d, col_r, col_d, br, bd, out, N);
}